// SLUTagging_61564061220928
// MI455X (gfx1250) — compile-verified
//
#include <hip/hip_runtime.h>

// SLUTagging Feat_Merge + tagging projection for MI455X (gfx1250).
//
//   feat[r,h] = out_word[b, word_idx[r], h]*rateW[r] + out_char[r,h]*rateC[r]
//   logits    = feat @ W_out + b_out     (fp32 via V_WMMA_F32_16X16X4_F32)
//
// Memory-bound (~290 MB @ 23.3 TB/s ~ 12.5 us floor, AI ~ 17 FLOP/B) ->
// keep full fp32 numerics; fp32 WMMA easily hides under the HBM stream.
// Padding columns (74..79) are computed with garbage B values from a
// clamped in-bounds address (column independence of D = A*B + C makes
// this safe) and discarded at the store — no exec-mask branching in the
// inner loop.

#define Bsz   16
#define Tsz   2048
#define Hsz   1024
#define NT    74          // num tags
#define ROWS  32          // rows per workgroup: 2 M-tiles per wave
#define KC    128         // K chunk staged in LDS per iteration
#define KPAD  (KC + 4)    // bank-stagger padding
#define WGT   160         // 5 wave32 per workgroup, one 16-wide N tile each
#define NCHUNK (Hsz / KC)
#define RATE_HEAD 0.88f
#define RATE_MID  0.70f

typedef __attribute__((ext_vector_type(2))) float v2f;
typedef __attribute__((ext_vector_type(8))) float v8f;

__launch_bounds__(WGT)
__global__ void slu_feat_gemm(const float* __restrict__ out_char,   // [B,T,H]
                              const float* __restrict__ out_word,   // [B,T,H]
                              const int*   __restrict__ word_idx,   // [B,T] int32
                              const unsigned char* __restrict__ is_head,    // [B,T] bool
                              const unsigned char* __restrict__ valid_mask, // [B,T] bool
                              const float* __restrict__ W,          // [H,NT]
                              const float* __restrict__ bias,       // [NT]
                              float* __restrict__ out)              // [B*T,NT]
{
    __shared__ float     s_feat[2][ROWS][KPAD];   // double-buffered blended tile
    __shared__ long long s_coff[ROWS];            // char row element offsets
    __shared__ long long s_woff[ROWS];            // gathered word row offsets
    __shared__ float     s_rw[ROWS];              // rate   (0 if invalid)
    __shared__ float     s_rc[ROWS];              // 1-rate (0 if invalid)

    const int tid  = threadIdx.x;
    const int lane = tid & 31;
    const int wave = tid >> 5;          // 0..4 -> 16-wide N tile
    const int n0   = wave * 16;
    const int row0 = blockIdx.x * ROWS; // flattened b*T + t

    // ---- per-row metadata: gather target + blend rates + validity ---------
    if (tid < ROWS) {
        const int r    = row0 + tid;
        const int b    = r >> 11;                    // T == 2048
        const int widx = word_idx[r];
        const float rate = is_head[r] ? RATE_HEAD : RATE_MID;
        const float v    = valid_mask[r] ? 1.0f : 0.0f;
        s_rw[tid]   = rate * v;
        s_rc[tid]   = (1.0f - rate) * v;
        s_coff[tid] = (long long)r * Hsz;
        s_woff[tid] = ((long long)b * Tsz + widx) * (long long)Hsz;
    }

    // WMMA 16x16x4 fp32 fragment coordinates (wave32):
    //   A: lane l, vgpr j -> A[M = l%16][K = 2*(l/16) + j]
    //   B: lane l, vgpr j -> B[K = 2*(l/16) + j][N = l%16]
    //   D: lane l, vgpr i -> D[M = i + 8*(l/16)][N = l%16]
    const int  m      = lane & 15;
    const int  khalf  = (lane >> 4) * 2;
    const int  n      = n0 + m;
    const bool nvalid = (n < NT);
    // Clamped, always-in-bounds W column base: padding lanes load real (but
    // discarded) data instead of taking exec-masked branches per load.
    const float* wcol = W + (nvalid ? n : 0);

    v8f acc0 = {};   // M-tile 0 (rows row0 +  0..15)
    v8f acc1 = {};   // M-tile 1 (rows row0 + 16..31)

    // ---- coalesced gather + rate-blend of one K chunk into LDS ------------
    auto stage = [&](int buf, int k0) {
        for (int idx = tid; idx < ROWS * (KC / 4); idx += WGT) {
            const int row = idx >> 5;            // KC/4 == 32 float4 per row
            const int c4  = idx & 31;
            const int k   = k0 + c4 * 4;
            const float4 cv = *reinterpret_cast<const float4*>(out_char + s_coff[row] + k);
            const float4 wv = *reinterpret_cast<const float4*>(out_word + s_woff[row] + k);
            const float  rw = s_rw[row];
            const float  rc = s_rc[row];
            float4 f;
            f.x = wv.x * rw + cv.x * rc;
            f.y = wv.y * rw + cv.y * rc;
            f.z = wv.z * rw + cv.z * rc;
            f.w = wv.w * rw + cv.w * rc;
            *reinterpret_cast<float4*>(&s_feat[buf][row][c4 * 4]) = f;
            if (k0 + KC < Hsz) {                 // warm L0/L2 for the chunk after
                __builtin_prefetch(out_char + s_coff[row] + k + KC, 0, 1);
                __builtin_prefetch(out_word + s_woff[row] + k + KC, 0, 1);
            }
        }
    };

    // ---- 32 fp32 WMMA K-steps over a staged chunk -------------------------
    auto compute = [&](int buf, int k0) {
        const float* wk = wcol + (size_t)(k0 + khalf) * NT;  // per-chunk base
        #pragma unroll 8
        for (int kk = 0; kk < KC; kk += 4) {
            v2f Bv;                                   // B[k..k+1][n], L2-hot
            Bv.x = wk[(size_t)kk * NT];
            Bv.y = wk[(size_t)(kk + 1) * NT];

            const float2 a0 = *reinterpret_cast<const float2*>(&s_feat[buf][m][kk + khalf]);
            const float2 a1 = *reinterpret_cast<const float2*>(&s_feat[buf][m + 16][kk + khalf]);
            v2f A0 = {a0.x, a0.y};
            v2f A1 = {a1.x, a1.y};

            acc0 = __builtin_amdgcn_wmma_f32_16x16x4_f32(
                       false, A0, false, Bv, (short)0, acc0, false, false);
            acc1 = __builtin_amdgcn_wmma_f32_16x16x4_f32(
                       false, A1, false, Bv, (short)0, acc1, false, false);
        }
    };

    // ---- software-pipelined main loop: stage(i+1) overlaps compute(i) -----
    __syncthreads();           // metadata visible
    stage(0, 0);
    __syncthreads();
    for (int kc = 0; kc < NCHUNK; ++kc) {
        const int cur = kc & 1;
        if (kc + 1 < NCHUNK) stage(cur ^ 1, (kc + 1) * KC);
        compute(cur, kc * KC);
        __syncthreads();
    }

    // ---- epilogue: bias + store (drop padding columns NT..79) -------------
    if (nvalid) {
        const float bb = bias[n];
        const int   mb = (lane >> 4) * 8;
        #pragma unroll
        for (int i = 0; i < 8; ++i) {
            out[(size_t)(row0 + mb + i) * NT + n]      = acc0[i] + bb;
            out[(size_t)(row0 + 16 + mb + i) * NT + n] = acc1[i] + bb;
        }
    }
}

extern "C" void kernel_launch(void* const* d_in, const int* in_sizes, int n_in,
                              void* d_out, int out_size, void* d_ws, size_t ws_size,
                              hipStream_t stream) {
    const float*         out_char   = (const float*)d_in[0];
    const float*         out_word   = (const float*)d_in[1];
    const int*           word_idx   = (const int*)d_in[2];
    const unsigned char* is_head    = (const unsigned char*)d_in[3];
    const unsigned char* valid_mask = (const unsigned char*)d_in[4];
    const float*         W          = (const float*)d_in[5];
    const float*         bias       = (const float*)d_in[6];
    float*               out        = (float*)d_out;

    const int rows = Bsz * Tsz;                  // 32768
    dim3 grid(rows / ROWS);                      // 1024 workgroups
    dim3 block(WGT);                             // 5 wave32 each
    hipLaunchKernelGGL(slu_feat_gemm, grid, block, 0, stream,
                       out_char, out_word, word_idx, is_head, valid_mask,
                       W, bias, out);
}